// WeightedAttentionPooling_10591389352019
// MI455X (gfx1250) — compile-verified
//
#include <hip/hip_runtime.h>
#include <hip/hip_bf16.h>
#include <math.h>

#define EPS 1e-10f

typedef __attribute__((ext_vector_type(2))) float v2f;
typedef __attribute__((ext_vector_type(8))) float v8f;

__device__ __forceinline__ float wave_reduce_sum(float v) {
#pragma unroll
  for (int off = 16; off > 0; off >>= 1) v += __shfl_xor(v, off, 32);
  return v;
}

__device__ __forceinline__ float wave_reduce_max(float v) {
#pragma unroll
  for (int off = 16; off > 0; off >>= 1) v = fmaxf(v, __shfl_xor(v, off, 32));
  return v;
}

// first position where a[pos] >= key (a sorted ascending)
__device__ __forceinline__ int lower_bound(const int* __restrict__ a, int n, int key) {
  int lo = 0, hi = n;
  while (lo < hi) {
    int mid = (lo + hi) >> 1;
    if (a[mid] < key) lo = mid + 1; else hi = mid;
  }
  return lo;
}

// ---------------------------------------------------------------------------
// Pass 1: gate[i] = dot(x[i,:], gate_w) + gate_b       (one wave per node)
// ---------------------------------------------------------------------------
__global__ void __launch_bounds__(256) gate_kernel(
    const float* __restrict__ x, const float* __restrict__ gate_w,
    const float* __restrict__ gate_b, float* __restrict__ gate, int N) {
  const int lane   = threadIdx.x & 31;
  const int wave   = (int)((blockIdx.x * blockDim.x + threadIdx.x) >> 5);
  const int nwaves = (int)((gridDim.x * blockDim.x) >> 5);
  const float4 gw = *(const float4*)(gate_w + lane * 4);
  const float  gb = gate_b[0];
  for (int node = wave; node < N; node += nwaves) {
    const float4 xv = *(const float4*)(x + (size_t)node * 128 + lane * 4);
    float s = xv.x * gw.x + xv.y * gw.y + xv.z * gw.z + xv.w * gw.w;
    s = wave_reduce_sum(s);
    if (lane == 0) gate[node] = s + gb;
  }
}

// ---------------------------------------------------------------------------
// Pass 2: per-segment softmax stats (one wave per segment, index is sorted).
// gate[i] is overwritten in place with c_i = w_i^p * exp(gate_i - max).
// inv[s] = 1/(sum+eps), sgate[s] = sum/(sum+eps). Pad segments -> 0 work.
// ---------------------------------------------------------------------------
__global__ void __launch_bounds__(256) seg_softmax_kernel(
    const int* __restrict__ index, const float* __restrict__ weights,
    const float* __restrict__ pow_p, float* __restrict__ gate,
    float* __restrict__ inv, float* __restrict__ sgate, int N, int S_pad) {
  const int lane = threadIdx.x & 31;
  const int seg  = (int)((blockIdx.x * blockDim.x + threadIdx.x) >> 5);
  if (seg >= S_pad) return;
  const int lo = lower_bound(index, N, seg);
  const int hi = lower_bound(index, N, seg + 1);
  const float p = pow_p[0];

  float m = -INFINITY;
  for (int i = lo + lane; i < hi; i += 32) m = fmaxf(m, gate[i]);
  m = wave_reduce_max(m);

  float sum = 0.0f;
  for (int i = lo + lane; i < hi; i += 32) {
    // weights in (0,1]; w^p = exp(p * ln w); fuse with exp(gate - max)
    float c = __expf(p * __logf(weights[i]) + gate[i] - m);
    gate[i] = c;
    sum += c;
  }
  sum = wave_reduce_sum(sum);
  if (lane == 0) {
    inv[seg]   = 1.0f / (sum + EPS);
    sgate[seg] = sum / (sum + EPS);
  }
}

// ---------------------------------------------------------------------------
// Pass 3: pooled[s,:] = inv[s] * sum_{i in seg s} c_i * x[i,:]
// One wave per segment; lanes stripe the 128-dim feature as float4.
// ---------------------------------------------------------------------------
__global__ void __launch_bounds__(256) pool_kernel(
    const int* __restrict__ index, const float* __restrict__ x,
    const float* __restrict__ coef, const float* __restrict__ inv,
    float* __restrict__ pooled, int N, int S_pad) {
  const int lane = threadIdx.x & 31;
  const int seg  = (int)((blockIdx.x * blockDim.x + threadIdx.x) >> 5);
  if (seg >= S_pad) return;
  const int lo = lower_bound(index, N, seg);
  const int hi = lower_bound(index, N, seg + 1);
  float4 acc = make_float4(0.f, 0.f, 0.f, 0.f);
  for (int i = lo; i < hi; ++i) {
    const float  c  = coef[i];
    const float4 xv = *(const float4*)(x + (size_t)i * 128 + lane * 4);
    acc.x += c * xv.x; acc.y += c * xv.y; acc.z += c * xv.z; acc.w += c * xv.w;
  }
  const float s = inv[seg];
  acc.x *= s; acc.y *= s; acc.z *= s; acc.w *= s;
  *(float4*)(pooled + (size_t)seg * 128 + lane * 4) = acc;
}

// ---------------------------------------------------------------------------
// Pass 4: out = pooled @ msg_w + sgate * msg_b  via V_WMMA_F32_16X16X4_F32.
// Block = 256 threads = 8 waves; block b -> M-tile b, wave w -> N-tile w.
// A (16x4, MxK) layout: lanes 0-15 hold K={k0,k0+1}, lanes 16-31 K={k0+2,k0+3}.
// B (4x16, KxN) layout: VGPR j holds row K=k0+j+(lanehalf*2), cols by lane&15.
// C/D (16x16)  layout: VGPR j holds row M=j+(lanehalf*8), cols by lane&15.
// ---------------------------------------------------------------------------
__global__ void __launch_bounds__(256) gemm_kernel(
    const float* __restrict__ pooled, const float* __restrict__ msg_w,
    const float* __restrict__ msg_b, const float* __restrict__ sgate,
    float* __restrict__ out, int S) {
  const int lane   = threadIdx.x & 31;
  const int tileN  = threadIdx.x >> 5;              // 0..7  (N = 128 = 8 tiles)
  const int tileM  = blockIdx.x;                    // 0..S_pad/16-1
  const int mrow   = tileM * 16 + (lane & 15);      // A-matrix row (padded, safe)
  const int ncol   = tileN * 16 + (lane & 15);      // B/C/D column
  const int khalf  = (lane >> 4) << 1;              // 0 or 2 per lane half

  const float* arow = pooled + (size_t)mrow * 128 + khalf;
  v8f c = {};
#pragma unroll 8
  for (int k0 = 0; k0 < 128; k0 += 4) {
    v2f a = *(const v2f*)(arow + k0);
    v2f b;
    b.x = msg_w[(size_t)(k0 + khalf) * 128 + ncol];
    b.y = msg_w[(size_t)(k0 + khalf + 1) * 128 + ncol];
    c = __builtin_amdgcn_wmma_f32_16x16x4_f32(false, a, false, b,
                                              (short)0, c, false, false);
  }

  const float bn    = msg_b[ncol];
  const int   rbase = tileM * 16 + ((lane >> 4) << 3);
#pragma unroll
  for (int j = 0; j < 8; ++j) {
    const int row = rbase + j;
    if (row < S) out[(size_t)row * 128 + ncol] = c[j] + sgate[row] * bn;
  }
}

// ---------------------------------------------------------------------------
extern "C" void kernel_launch(void* const* d_in, const int* in_sizes, int n_in,
                              void* d_out, int out_size, void* d_ws, size_t ws_size,
                              hipStream_t stream) {
  (void)n_in; (void)ws_size;
  const float* x       = (const float*)d_in[0];
  const int*   index   = (const int*)  d_in[1];
  const float* weights = (const float*)d_in[2];
  const float* gate_w  = (const float*)d_in[3];
  const float* gate_b  = (const float*)d_in[4];
  const float* msg_w   = (const float*)d_in[5];
  const float* msg_b   = (const float*)d_in[6];
  const float* pow_p   = (const float*)d_in[7];
  // d_in[8] = num_segments lives on device; derive S on host from out_size.

  const int D     = 128;
  const int N     = in_sizes[0] / D;
  const int S     = out_size / D;
  const int S_pad = (S + 15) & ~15;

  float* ws     = (float*)d_ws;
  float* gate   = ws;                       // N floats (becomes coef in-place)
  float* inv    = ws + N;                   // S_pad floats
  float* sg     = ws + N + S_pad;           // S_pad floats
  float* pooled = ws + N + 2 * S_pad;       // S_pad*128 floats (16B aligned)

  float* out = (float*)d_out;

  gate_kernel<<<2048, 256, 0, stream>>>(x, gate_w, gate_b, gate, N);

  const int segBlocks = (S_pad * 32 + 255) / 256;   // one wave per segment
  seg_softmax_kernel<<<segBlocks, 256, 0, stream>>>(index, weights, pow_p,
                                                    gate, inv, sg, N, S_pad);
  pool_kernel<<<segBlocks, 256, 0, stream>>>(index, x, gate, inv, pooled, N, S_pad);

  gemm_kernel<<<S_pad / 16, 256, 0, stream>>>(pooled, msg_w, msg_b, sg, out, S);
}